// CornerAwareDCN_74603581931895
// MI455X (gfx1250) — compile-verified
//
#include <hip/hip_runtime.h>

#define HH 256
#define WW 256
#define HW (HH*WW)
#define DIMC 64
#define NB 2

typedef __attribute__((ext_vector_type(2))) float v2f;
typedef __attribute__((ext_vector_type(8))) float v8f;

__device__ __forceinline__ float lrelu(float v){ return v >= 0.f ? v : 0.2f*v; }
__device__ __forceinline__ float sigmoidf(float v){ return 1.f/(1.f + expf(-v)); }

// ---------------------------------------------------------------------------
// Stage 1: corner prior  cp = sigmoid(pw(lrelu(dw3x3(x))))
// ---------------------------------------------------------------------------
__global__ void cp_kernel(const float* __restrict__ x, const float* __restrict__ dww,
                          const float* __restrict__ pww, const float* __restrict__ pwb,
                          float* __restrict__ cp){
  int idx = blockIdx.x*blockDim.x + threadIdx.x;
  if (idx >= NB*HW) return;
  int b = idx / HW, p = idx % HW;
  int y = p / WW, xx = p % WW;
  float acc = pwb[0];
  for (int cin = 0; cin < DIMC; ++cin){
    const float* xp = x + ((size_t)(b*DIMC + cin))*HW;
    float s = 0.f;
#pragma unroll
    for (int t = 0; t < 9; ++t){
      int dy = t/3 - 1, dx = t%3 - 1;
      int yy = y + dy, xq = xx + dx;
      if (yy >= 0 && yy < HH && xq >= 0 && xq < WW)
        s += dww[cin*9 + t] * xp[yy*WW + xq];
    }
    acc += pww[cin] * lrelu(s);
  }
  cp[idx] = sigmoidf(acc);
}

// ---------------------------------------------------------------------------
// Generic implicit-GEMM conv (3x3 SAME or 1x1) via V_WMMA_F32_16X16X4_F32.
// Tap-major K ordering (K = tap*Cpad + cin): no per-element divisions, tap
// geometry hoisted. One wave computes MT 16(out-ch)x16(pixel) tiles, reusing
// each B fragment MT times.
// ---------------------------------------------------------------------------
struct ConvArgs {
  const float* seg[5];
  int          segc[5];
  const float* wt;     // (O, Ctot, k, k) flat -> o*Ktot + cin*taps + tap
  const float* bias;   // (O)
  const float* res0;   // optional residual, first res_split channels
  const float* res1;   // optional residual, remaining O-res_split channels
  int          res_split;
  float*       out;    // (B, O, H, W)
  int O, Ctot, ksz, act;
};

template<int MT>
__global__ __launch_bounds__(128) void conv_wmma(ConvArgs a){
  const int tid  = threadIdx.x;
  const int wave = tid >> 5, lane = tid & 31;
  const int n  = lane & 15, hi = lane >> 4;
  const int y  = blockIdx.y;
  const int b  = blockIdx.z;
  const int x0 = (blockIdx.x*4 + wave)*16;
  const int taps = a.ksz * a.ksz;            // 9 or 1
  const int Ktot = a.Ctot * taps;
  const int Cpad = (a.Ctot + 3) & ~3;

  v8f acc[MT];
#pragma unroll
  for (int m = 0; m < MT; ++m) acc[m] = (v8f){};

  for (int tap = 0; tap < taps; ++tap){
    const int dy = (taps == 9) ? (tap/3 - 1) : 0;
    const int dx = (taps == 9) ? (tap%3 - 1) : 0;
    const int yy = y + dy;
    const int xx = x0 + n + dx;
    const bool pix_ok = (yy >= 0) & (yy < HH) & (xx >= 0) & (xx < WW);
    const size_t poff = (size_t)yy*WW + xx;

    for (int cc = 0; cc < Cpad; cc += 4){
      v2f bv;
#pragma unroll
      for (int c = 0; c < 2; ++c){
        int cin = cc + 2*hi + c;             // per-lane K slot (ISA A/B layout)
        float bval = 0.f;
        if (pix_ok && cin < a.Ctot){
          int cl = cin; const float* sp = nullptr; int sc = 0;
#pragma unroll
          for (int s = 0; s < 5; ++s){
            int cseg = a.segc[s];
            if (sp == nullptr){
              if (cl < cseg){ sp = a.seg[s]; sc = cseg; } else cl -= cseg;
            }
          }
          if (sp) bval = sp[((size_t)(b*sc + cl))*HW + poff];
        }
        bv[c] = bval;
      }
#pragma unroll
      for (int m = 0; m < MT; ++m){
        const int oc = m*16 + n;
        v2f av;
#pragma unroll
        for (int c = 0; c < 2; ++c){
          int cin = cc + 2*hi + c;
          av[c] = (oc < a.O && cin < a.Ctot)
                  ? a.wt[(size_t)oc*Ktot + cin*taps + tap] : 0.f;
        }
        acc[m] = __builtin_amdgcn_wmma_f32_16x16x4_f32(false, av, false, bv,
                                                       (short)0, acc[m], false, false);
      }
    }
  }
  // epilogue: ISA D layout: M = v + 8*hi, N = n
#pragma unroll
  for (int m = 0; m < MT; ++m){
#pragma unroll
    for (int v = 0; v < 8; ++v){
      int oc = m*16 + v + 8*hi;
      if (oc < a.O){
        float val = acc[m][v] + a.bias[oc];
        if (a.act) val = lrelu(val);
        size_t p = (size_t)y*WW + (x0 + n);
        if (a.res0){
          float r;
          if (oc < a.res_split) r = a.res0[((size_t)(b*a.res_split + oc))*HW + p];
          else r = a.res1[((size_t)(b*(a.O - a.res_split) + (oc - a.res_split)))*HW + p];
          val += r;
        }
        a.out[((size_t)(b*a.O + oc))*HW + p] = val;
      }
    }
  }
}

// ---------------------------------------------------------------------------
// Stage 3: per-(b,tap,pixel) bilinear sampling tables from offsets+mask.
// Weights fold in corner validity and sigmoid(mask).
// ---------------------------------------------------------------------------
__global__ void samp_kernel(const float* __restrict__ om, float* __restrict__ sw,
                            int* __restrict__ si){
  int idx = blockIdx.x*blockDim.x + threadIdx.x;
  if (idx >= NB*9*HW) return;
  int b = idx / (9*HW);
  int r = idx % (9*HW);
  int tap = r / HW;
  int p = r % HW;
  int y = p / WW, x = p % WW;
  const float* omb = om + (size_t)b*27*HW;
  float offy = omb[(size_t)(tap*2+0)*HW + p];
  float offx = omb[(size_t)(tap*2+1)*HW + p];
  float m    = sigmoidf(omb[(size_t)(18+tap)*HW + p]);
  float py = (float)y + (float)(tap/3 - 1) + offy;
  float px = (float)x + (float)(tap%3 - 1) + offx;
  float fy0 = floorf(py), fx0 = floorf(px);
  float ly = py - fy0, lx = px - fx0;
  float fy1 = fy0 + 1.f, fx1 = fx0 + 1.f;
  float vy0 = (fy0 >= 0.f && fy0 < (float)HH) ? 1.f : 0.f;
  float vy1 = (fy1 >= 0.f && fy1 < (float)HH) ? 1.f : 0.f;
  float vx0 = (fx0 >= 0.f && fx0 < (float)WW) ? 1.f : 0.f;
  float vx1 = (fx1 >= 0.f && fx1 < (float)WW) ? 1.f : 0.f;
  sw[(size_t)idx*4+0] = (1.f-ly)*(1.f-lx) * vy0*vx0 * m;
  sw[(size_t)idx*4+1] = (1.f-ly)*lx       * vy0*vx1 * m;
  sw[(size_t)idx*4+2] = ly*(1.f-lx)       * vy1*vx0 * m;
  sw[(size_t)idx*4+3] = ly*lx             * vy1*vx1 * m;
  si[(size_t)idx*2+0] = (int)fminf(fmaxf(fy0, 0.f), (float)(HH-1));
  si[(size_t)idx*2+1] = (int)fminf(fmaxf(fx0, 0.f), (float)(WW-1));
}

// ---------------------------------------------------------------------------
// Stage 4: deformable conv as implicit GEMM (M=64, K=9*64) with WMMA.
// Tap-major K ordering: sampling table read once per tap (8 LDS values),
// reused across all 16 cin-steps; each gathered B fragment feeds 4 WMMAs.
// out = x + gamma*(acc + bias).
// ---------------------------------------------------------------------------
__global__ __launch_bounds__(128) void dcn_wmma(const float* __restrict__ x,
                                                const float* __restrict__ sw,
                                                const int* __restrict__ si,
                                                const float* __restrict__ wt,
                                                const float* __restrict__ bias,
                                                const float* __restrict__ gamma,
                                                float* __restrict__ out){
  __shared__ float sWt[9*64*4];
  __shared__ int   sIx[9*64*4];
  const int tid  = threadIdx.x;
  const int wave = tid >> 5, lane = tid & 31;
  const int n  = lane & 15, hi = lane >> 4;
  const int y  = blockIdx.y;
  const int b  = blockIdx.z;
  const int xblk = blockIdx.x * 64;

  for (int t = tid; t < 9*64; t += 128){
    int tap = t >> 6, pxl = t & 63;
    size_t e = ((size_t)(b*9 + tap))*HW + (size_t)y*WW + (xblk + pxl);
#pragma unroll
    for (int j = 0; j < 4; ++j) sWt[t*4+j] = sw[e*4+j];
    int cy0 = si[e*2+0], cx0 = si[e*2+1];
    int cy1 = cy0+1 < HH ? cy0+1 : HH-1;
    int cx1 = cx0+1 < WW ? cx0+1 : WW-1;
    sIx[t*4+0] = cy0*WW + cx0;
    sIx[t*4+1] = cy0*WW + cx1;
    sIx[t*4+2] = cy1*WW + cx0;
    sIx[t*4+3] = cy1*WW + cx1;
  }
  __syncthreads();

  const int x0 = xblk + wave*16;
  const int lpix = wave*16 + n;
  const float* xb = x + (size_t)b*DIMC*HW;

  v8f acc[4];
#pragma unroll
  for (int m = 0; m < 4; ++m) acc[m] = (v8f){};

#pragma unroll
  for (int tap = 0; tap < 9; ++tap){
    const int t = (tap<<6) + lpix;
    const float w0 = sWt[t*4+0], w1 = sWt[t*4+1], w2 = sWt[t*4+2], w3 = sWt[t*4+3];
    const int   i0 = sIx[t*4+0], i1 = sIx[t*4+1], i2 = sIx[t*4+2], i3 = sIx[t*4+3];
    for (int cc = 0; cc < DIMC; cc += 4){
      v2f bv;
#pragma unroll
      for (int c = 0; c < 2; ++c){
        const float* xp = xb + (size_t)(cc + 2*hi + c)*HW;
        bv[c] = w0*xp[i0] + w1*xp[i1] + w2*xp[i2] + w3*xp[i3];
      }
#pragma unroll
      for (int m = 0; m < 4; ++m){
        v2f av;
#pragma unroll
        for (int c = 0; c < 2; ++c)
          av[c] = wt[(size_t)(m*16 + n)*576 + (cc + 2*hi + c)*9 + tap];
        acc[m] = __builtin_amdgcn_wmma_f32_16x16x4_f32(false, av, false, bv,
                                                       (short)0, acc[m], false, false);
      }
    }
  }
#pragma unroll
  for (int m = 0; m < 4; ++m){
#pragma unroll
    for (int v = 0; v < 8; ++v){
      int oc = m*16 + v + 8*hi;
      size_t o = ((size_t)(b*DIMC + oc))*HW + (size_t)y*WW + (x0 + n);
      out[o] = x[o] + gamma[oc]*(acc[m][v] + bias[oc]);
    }
  }
}

// ---------------------------------------------------------------------------
extern "C" void kernel_launch(void* const* d_in, const int* in_sizes, int n_in,
                              void* d_out, int out_size, void* d_ws, size_t ws_size,
                              hipStream_t stream){
  const float* x     = (const float*)d_in[0];
  const float* cpdw  = (const float*)d_in[1];
  const float* cppw  = (const float*)d_in[2];
  const float* cppb  = (const float*)d_in[3];
  const float* c1w   = (const float*)d_in[4];
  const float* c1b   = (const float*)d_in[5];
  const float* c2w   = (const float*)d_in[6];
  const float* c2b   = (const float*)d_in[7];
  const float* c3w   = (const float*)d_in[8];
  const float* c3b   = (const float*)d_in[9];
  const float* fw    = (const float*)d_in[10];
  const float* fb    = (const float*)d_in[11];
  const float* og1w  = (const float*)d_in[12];
  const float* og1b  = (const float*)d_in[13];
  const float* og2w  = (const float*)d_in[14];
  const float* og2b  = (const float*)d_in[15];
  const float* dcnw  = (const float*)d_in[16];
  const float* dcnb  = (const float*)d_in[17];
  const float* gam   = (const float*)d_in[18];
  float* out = (float*)d_out;

  float* ws = (float*)d_ws;
  size_t o = 0;
  float* cp   = ws + o; o += (size_t)NB*HW;
  float* x1   = ws + o; o += (size_t)NB*16*HW;
  float* x2   = ws + o; o += (size_t)NB*16*HW;
  float* x3   = ws + o; o += (size_t)NB*16*HW;
  float* rdb  = ws + o; o += (size_t)NB*65*HW;
  float* og1o = ws + o; o += (size_t)NB*32*HW;
  float* om   = ws + o; o += (size_t)NB*27*HW;
  float* swp  = ws + o; o += (size_t)NB*9*HW*4;
  int*   sip  = (int*)(ws + o);

  // Stage 1: corner prior
  {
    int tot = NB*HW;
    cp_kernel<<<(tot+255)/256, 256, 0, stream>>>(x, cpdw, cppw, cppb, cp);
  }

  dim3 blk(128);
  dim3 grd(WW/64, HH, NB);

  // c1: cat(x, cp) 65 -> 16, lrelu
  { ConvArgs a{}; a.seg[0]=x; a.segc[0]=64; a.seg[1]=cp; a.segc[1]=1;
    a.wt=c1w; a.bias=c1b; a.out=x1; a.O=16; a.Ctot=65; a.ksz=3; a.act=1;
    conv_wmma<1><<<grd, blk, 0, stream>>>(a); }
  // c2: cat(xc, x1) 81 -> 16, lrelu
  { ConvArgs a{}; a.seg[0]=x; a.segc[0]=64; a.seg[1]=cp; a.segc[1]=1;
    a.seg[2]=x1; a.segc[2]=16;
    a.wt=c2w; a.bias=c2b; a.out=x2; a.O=16; a.Ctot=81; a.ksz=3; a.act=1;
    conv_wmma<1><<<grd, blk, 0, stream>>>(a); }
  // c3: cat(xc, x1, x2) 97 -> 16, lrelu
  { ConvArgs a{}; a.seg[0]=x; a.segc[0]=64; a.seg[1]=cp; a.segc[1]=1;
    a.seg[2]=x1; a.segc[2]=16; a.seg[3]=x2; a.segc[3]=16;
    a.wt=c3w; a.bias=c3b; a.out=x3; a.O=16; a.Ctot=97; a.ksz=3; a.act=1;
    conv_wmma<1><<<grd, blk, 0, stream>>>(a); }
  // fuse: 1x1 cat(xc,x1,x2,x3) 113 -> 65, + xc residual
  { ConvArgs a{}; a.seg[0]=x; a.segc[0]=64; a.seg[1]=cp; a.segc[1]=1;
    a.seg[2]=x1; a.segc[2]=16; a.seg[3]=x2; a.segc[3]=16; a.seg[4]=x3; a.segc[4]=16;
    a.wt=fw; a.bias=fb; a.out=rdb; a.O=65; a.Ctot=113; a.ksz=1; a.act=0;
    a.res0=x; a.res1=cp; a.res_split=64;
    conv_wmma<5><<<grd, blk, 0, stream>>>(a); }
  // og1: 65 -> 32, lrelu
  { ConvArgs a{}; a.seg[0]=rdb; a.segc[0]=65;
    a.wt=og1w; a.bias=og1b; a.out=og1o; a.O=32; a.Ctot=65; a.ksz=3; a.act=1;
    conv_wmma<2><<<grd, blk, 0, stream>>>(a); }
  // og2: 32 -> 27 (offsets + mask logits)
  { ConvArgs a{}; a.seg[0]=og1o; a.segc[0]=32;
    a.wt=og2w; a.bias=og2b; a.out=om; a.O=27; a.Ctot=32; a.ksz=3; a.act=0;
    conv_wmma<2><<<grd, blk, 0, stream>>>(a); }

  // Stage 3: bilinear sampling tables
  {
    int tot = NB*9*HW;
    samp_kernel<<<(tot+255)/256, 256, 0, stream>>>(om, swp, sip);
  }
  // Stage 4: deformable conv GEMM + residual
  dcn_wmma<<<grd, blk, 0, stream>>>(x, swp, sip, dcnw, dcnb, gam, out);
}